// GDL_CrossEntropy_47931835023728
// MI455X (gfx1250) — compile-verified
//
#include <hip/hip_runtime.h>
#include <hip/hip_bf16.h>
#include <stdint.h>

// ---------------------------------------------------------------------------
// GDL + CrossEntropy fused loss for MI455X (gfx1250, wave32)
// logits [16,6,512,512] f32, labels [16,512,512] i64 -> scalar f32
// Memory-bound: ~134 MB single-use traffic -> ~5.8 us floor @ 23.3 TB/s.
// ---------------------------------------------------------------------------

#define HW (512 * 512)
#define NC 6
#define NB 16
#define PIX_TOTAL (NB * HW)        // 4194304
// ws layout (floats): [0]=ce_sum, [1..96]=pred[b*6+c], [97..192]=inter, [193..288]=gcount
#define WS_PRED  1
#define WS_INTER 97
#define WS_CNT   193
#define WS_N     289

typedef float  __attribute__((ext_vector_type(2))) v2f;
typedef float  __attribute__((ext_vector_type(4))) f32x4;
typedef float  __attribute__((ext_vector_type(8))) v8f;
typedef int    __attribute__((ext_vector_type(4))) i32x4;

__global__ __launch_bounds__(320) void gdl_zero(float* __restrict__ ws) {
    int i = threadIdx.x;
    if (i < WS_N) ws[i] = 0.0f;
}

// grid = (256, 16), block = 256. Block handles 1024 consecutive pixels of batch b.
__global__ __launch_bounds__(256) void gdl_main(const float* __restrict__ logits,
                                                const long long* __restrict__ labels,
                                                float* __restrict__ ws) {
    __shared__ float tile[NC * 1024];        // 24 KB: 6 class rows x 1024 pixels
    __shared__ float partials[8 * 19];

    const int tid  = threadIdx.x;            // 0..255
    const int lane = tid & 31;
    const int wave = tid >> 5;               // 0..7 (wave32)
    const int b    = blockIdx.y;
    const int p0   = blockIdx.x * 1024;

    // --- async global -> LDS streaming of the 6 class rows (b128 per lane) ---
    const float* gbase = logits + (size_t)b * NC * HW + p0 + tid * 4;
#pragma unroll
    for (int c = 0; c < NC; ++c) {
        uint32_t laddr = (uint32_t)(uintptr_t)(&tile[c * 1024 + tid * 4]);
        uint64_t gaddr = (uint64_t)(uintptr_t)(gbase + (size_t)c * HW);
        asm volatile("global_load_async_to_lds_b128 %0, %1, off"
                     :: "v"(laddr), "v"(gaddr) : "memory");
    }

    // --- labels: direct nontemporal b128 loads (4 x i64 = 32B) ---
    const i32x4* lp = (const i32x4*)(labels + (size_t)b * HW + p0 + (size_t)tid * 4);
    i32x4 q0 = __builtin_nontemporal_load(lp);
    i32x4 q1 = __builtin_nontemporal_load(lp + 1);
    int lab[4] = { q0.x, q0.z, q1.x, q1.z };  // low dword of each little-endian i64

    asm volatile("s_wait_asynccnt 0x0" ::: "memory");

    // Each thread reads back exactly the bytes its own async ops wrote -> no barrier.
    float l[NC][4];
#pragma unroll
    for (int c = 0; c < NC; ++c) {
        f32x4 v = *(const f32x4*)(&tile[c * 1024 + tid * 4]);
        l[c][0] = v.x; l[c][1] = v.y; l[c][2] = v.z; l[c][3] = v.w;
    }

    float ce = 0.0f;
    float pred[NC]  = {0.f, 0.f, 0.f, 0.f, 0.f, 0.f};
    float inter[NC] = {0.f, 0.f, 0.f, 0.f, 0.f, 0.f};
    float cnt[NC]   = {0.f, 0.f, 0.f, 0.f, 0.f, 0.f};

#pragma unroll
    for (int p = 0; p < 4; ++p) {
        float m = l[0][p];
#pragma unroll
        for (int c = 1; c < NC; ++c) m = fmaxf(m, l[c][p]);
        float e[NC], s = 0.0f;
#pragma unroll
        for (int c = 0; c < NC; ++c) { e[c] = __expf(l[c][p] - m); s += e[c]; }
        const float inv = 1.0f / s;
        const int lb = lab[p];
        float lsel = 0.0f;
#pragma unroll
        for (int c = 0; c < NC; ++c) {
            float pc = e[c] * inv;
            pred[c] += pc;
            bool hit = (lb == c);
            if (hit) { inter[c] += pc; cnt[c] += 1.0f; }
            lsel = hit ? l[c][p] : lsel;
        }
        if (lb >= 0) ce += m + __logf(s) - lsel;   // -log p[label]; ignored -> 0
    }

    // --- wave32 butterfly reduce of the 19 accumulators ---
    float red[19];
    red[0] = ce;
#pragma unroll
    for (int c = 0; c < NC; ++c) {
        red[1 + c]  = pred[c];
        red[7 + c]  = inter[c];
        red[13 + c] = cnt[c];
    }
#pragma unroll
    for (int j = 0; j < 19; ++j) {
        float v = red[j];
#pragma unroll
        for (int off = 16; off > 0; off >>= 1) v += __shfl_xor(v, off, 32);
        if (lane == 0) partials[wave * 19 + j] = v;
    }
    __syncthreads();

    // --- cross-wave reduce + one atomicAdd set per block (19 atomics) ---
    if (tid < 19) {
        float s = 0.0f;
#pragma unroll
        for (int w = 0; w < 8; ++w) s += partials[w * 19 + tid];
        int addr;
        if (tid == 0)       addr = 0;
        else if (tid < 7)   addr = WS_PRED  + b * NC + (tid - 1);
        else if (tid < 13)  addr = WS_INTER + b * NC + (tid - 7);
        else                addr = WS_CNT   + b * NC + (tid - 13);
        atomicAdd(&ws[addr], s);
    }
}

// 1 block of 32 threads. Batch-dim (16) reduction of 2*inter*w (cols 0..5) and
// (g+pred)*w (cols 6..11) via chained V_WMMA_F32_16X16X4_F32 with A = ones.
// B values are staged branchlessly into a dense 16x16 LDS matrix first, so the
// WMMA feed is straight ds_load_b32 with EXEC all-ones (ISA 7.12 requirement).
__global__ __launch_bounds__(32) void gdl_finalize(const float* __restrict__ ws,
                                                   float* __restrict__ out) {
    __shared__ float sg[96], spred[96], sinter[96], sw[96];
    __shared__ float bval[NB * 16];   // [b][n]: n 0..5 numer, 6..11 denom, 12..15 zero
    __shared__ float res[32];
    const int tid = threadIdx.x;      // 0..31, all lanes active

    for (int i = tid; i < 96; i += 32) {      // 3 uniform iterations
        float g   = ws[WS_CNT + i];
        sg[i]     = g;
        spred[i]  = ws[WS_PRED + i];
        sinter[i] = ws[WS_INTER + i];
        sw[i]     = (g > 0.0f) ? 1.0f / (g * g) : 0.0f;   // inf entries zeroed
    }
    __syncthreads();
    if (tid < NB) {   // per-batch-row inf fixup: inf -> max of finite row entries
        float mx = 0.0f;
#pragma unroll
        for (int c = 0; c < NC; ++c) mx = fmaxf(mx, sw[tid * NC + c]);
#pragma unroll
        for (int c = 0; c < NC; ++c)
            sw[tid * NC + c] = (sg[tid * NC + c] == 0.0f) ? mx : sw[tid * NC + c];
    }
    __syncthreads();

    // Branchless staging of the WMMA B-matrix values (8 uniform iterations).
    for (int i = tid; i < NB * 16; i += 32) {
        int bb = i >> 4;
        int n  = i & 15;
        int c  = (n < NC) ? n : ((n < 2 * NC) ? n - NC : 0);
        int k  = bb * NC + c;
        float wv    = sw[k];
        float numer = 2.0f * sinter[k] * wv;
        float denom = (sg[k] + spred[k]) * wv;
        float v     = (n < NC) ? numer : denom;
        bval[i]     = (n < 2 * NC) ? v : 0.0f;
    }
    __syncthreads();

#if __has_builtin(__builtin_amdgcn_wmma_f32_16x16x4_f32)
    {
        const int n  = tid & 15;     // B/D column
        const int hi = tid >> 4;     // lane half selects K pair {2hi, 2hi+1}
        v8f acc = {0.f, 0.f, 0.f, 0.f, 0.f, 0.f, 0.f, 0.f};
        v2f a;  a.x = 1.0f; a.y = 1.0f;       // A = ones: D[m][n] = sum_k B[k][n]
#pragma unroll
        for (int j = 0; j < 4; ++j) {         // K=4 per op, 4 ops reduce all 16 batches
            v2f bm;
            bm.x = bval[(j * 4 + hi * 2 + 0) * 16 + n];
            bm.y = bval[(j * 4 + hi * 2 + 1) * 16 + n];
            acc = __builtin_amdgcn_wmma_f32_16x16x4_f32(
                false, a, false, bm, (short)0, acc, false, false);
        }
        res[tid] = acc[0];                    // row 0 (all rows identical)
    }
#else
    if (tid == 0) {
        for (int n = 0; n < 12; ++n) {
            float s = 0.0f;
            for (int bb = 0; bb < NB; ++bb) s += bval[bb * 16 + n];
            res[n] = s;
        }
    }
#endif
    __syncthreads();

    if (tid == 0) {
        float fsum = 0.0f;
#pragma unroll
        for (int c = 0; c < NC; ++c)
            fsum += 1.0f - (res[c] + 1e-5f) / (res[c + NC] + 1e-5f);
        out[0] = fsum / (float)NC + ws[0] / (float)PIX_TOTAL;
    }
}

extern "C" void kernel_launch(void* const* d_in, const int* in_sizes, int n_in,
                              void* d_out, int out_size, void* d_ws, size_t ws_size,
                              hipStream_t stream) {
    (void)in_sizes; (void)n_in; (void)out_size; (void)ws_size;
    const float*     logits = (const float*)d_in[0];
    const long long* labels = (const long long*)d_in[1];
    float* ws  = (float*)d_ws;
    float* out = (float*)d_out;

    gdl_zero<<<1, 320, 0, stream>>>(ws);
    gdl_main<<<dim3(256, 16), 256, 0, stream>>>(logits, labels, ws);
    gdl_finalize<<<1, 32, 0, stream>>>(ws, out);
}